// GraphNetwork_23922967838770
// MI455X (gfx1250) — compile-verified
//
#include <hip/hip_runtime.h>

typedef __attribute__((ext_vector_type(2))) float v2f;
typedef __attribute__((ext_vector_type(8))) float v8f;

#define N_NODES 100000
#define N_EDGES 1600000
#define F_IN    256
#define F_OUT   64
#define N_REL   2
#define N_BASES 30
#define N_TILES (N_NODES / 16)      // 6250 (exact)
#define NQ      (F_IN / 8)          // 32 K-octets
#define WPB     16                  // waves per block in GEMM kernel
#define W_FLOATS (3 * F_IN * F_OUT) // 49152 floats = 196608 B (fits 320KB LDS)

// ---------------------------------------------------------------------------
// Kernel 0: fold att@basis (+root) into K-octet packed layout:
//   P[ ((r*NQ + q)*F_OUT + n)*8 + h*4 + s*2 + j ] = W[r][8q + 4s + 2h + j][n]
// A lane (half h, column n) reads the B fragments of one octet as two
// adjacent 8-byte pairs (s=0 -> kb, s=1 -> kb+4): one ds_load_2addr_b64.
// ---------------------------------------------------------------------------
__global__ void build_weights(const float* __restrict__ basis,
                              const float* __restrict__ att,
                              const float* __restrict__ root,
                              float* __restrict__ P) {
    int idx = blockIdx.x * blockDim.x + threadIdx.x;  // over 3*256*64
    if (idx >= W_FLOATS) return;
    int n = idx % F_OUT;
    int k = (idx / F_OUT) % F_IN;
    int r = idx / (F_OUT * F_IN);
    float w;
    if (r < N_REL) {
        w = 0.0f;
        #pragma unroll
        for (int b = 0; b < N_BASES; ++b)
            w += att[r * N_BASES + b] * basis[(b * F_IN + k) * F_OUT + n];
    } else {
        w = root[k * F_OUT + n];
    }
    const int q = k >> 3;
    const int s = (k >> 2) & 1;
    const int h = (k >> 1) & 1;
    const int j = k & 1;
    P[((r * NQ + q) * F_OUT + n) * 8 + h * 4 + s * 2 + j] = w;
}

// ---------------------------------------------------------------------------
// Kernel 1: WMMA GEMM. Each wave: one 16-row node tile, all 64 output cols,
// 3 weight matrices (rel0, rel1, root). f32 16x16x4 WMMA.
// Fully unrolled K loop (scheduler keeps many DS loads in flight with
// partial s_wait_dscnt); B pairs merge into ds_load_2addr_b64.
// ---------------------------------------------------------------------------
__global__ void __launch_bounds__(32 * WPB)
gemm_wmma(const float* __restrict__ x, const float* __restrict__ P,
          const float* __restrict__ bias,
          float* __restrict__ h, float* __restrict__ out) {
    extern __shared__ float sW[];  // W_FLOATS floats, octet-packed layout

    // cooperative stage of packed weights into LDS (16B chunks)
    {
        const float4* __restrict__ Pv = (const float4*)P;
        float4* sWv = (float4*)sW;
        for (int i = threadIdx.x; i < W_FLOATS / 4; i += blockDim.x)
            sWv[i] = Pv[i];
    }
    __syncthreads();

    const int wave = threadIdx.x >> 5;
    const int lane = threadIdx.x & 31;
    const int tile = blockIdx.x * WPB + wave;
    if (tile >= N_TILES) return;

    const int half = lane >> 4;   // A: K{0,1} vs K{2,3}; B: same split
    const int lm   = lane & 15;   // A: row M;  B/C/D: column N (within tile)

    v8f acc[3][4];
    #pragma unroll
    for (int r = 0; r < 3; ++r)
        #pragma unroll
        for (int t = 0; t < 4; ++t) {
            v8f z = {};
            acc[r][t] = z;
        }

    // Per-relation lane base pointers (each relation slice is exactly 64KB)
    const float* bp0 = sW + 0 * NQ * F_OUT * 8 + lm * 8 + half * 4;
    const float* bp1 = sW + 1 * NQ * F_OUT * 8 + lm * 8 + half * 4;
    const float* bp2 = sW + 2 * NQ * F_OUT * 8 + lm * 8 + half * 4;

    // A fragment base: row = tile*16 + lm, per-lane K offset half*2
    const float* xptr = x + (size_t)(tile * 16 + lm) * F_IN + half * 2;

    #pragma unroll
    for (int q = 0; q < NQ; ++q) {
        // A fragments for the two K-steps of this octet
        v2f a0 = *(const v2f*)(xptr + 8 * q);       // kb = 8q
        v2f a1 = *(const v2f*)(xptr + 8 * q + 4);   // kb = 8q + 4

        // batch all B-fragment pairs into distinct regs
        v2f blo[12], bhi[12];
        #pragma unroll
        for (int t = 0; t < 4; ++t) {
            const int o = q * (F_OUT * 8) + t * 128;  // float offset
            blo[0 + t] = *(const v2f*)(bp0 + o);
            bhi[0 + t] = *(const v2f*)(bp0 + o + 2);
            blo[4 + t] = *(const v2f*)(bp1 + o);
            bhi[4 + t] = *(const v2f*)(bp1 + o + 2);
            blo[8 + t] = *(const v2f*)(bp2 + o);
            bhi[8 + t] = *(const v2f*)(bp2 + o + 2);
        }

        #pragma unroll
        for (int i = 0; i < 12; ++i) {
            const int r = i >> 2, t = i & 3;
            acc[r][t] = __builtin_amdgcn_wmma_f32_16x16x4_f32(
                false, a0, false, blo[i], (short)0, acc[r][t], false, false);
            acc[r][t] = __builtin_amdgcn_wmma_f32_16x16x4_f32(
                false, a1, false, bhi[i], (short)0, acc[r][t], false, false);
        }
    }

    // D layout (32-bit 16x16): lane<16 -> N=lane, M=v; lane>=16 -> N=lane-16, M=v+8
    const int node0 = tile * 16;
    #pragma unroll
    for (int t = 0; t < 4; ++t) {
        const int col = t * 16 + lm;
        const float bv = bias[col];
        #pragma unroll
        for (int v = 0; v < 8; ++v) {
            const int node = node0 + v + half * 8;
            h[(size_t)(0 * N_NODES + node) * F_OUT + col] = acc[0][t][v];
            h[(size_t)(1 * N_NODES + node) * F_OUT + col] = acc[1][t][v];
            out[(size_t)node * F_OUT + col] = acc[2][t][v] + bv;
        }
    }
}

// ---------------------------------------------------------------------------
// Kernel 2: zero agg + cnt (contiguous region, float4 stores)
// ---------------------------------------------------------------------------
__global__ void zero_ws(float4* __restrict__ p, int n4) {
    int i = blockIdx.x * blockDim.x + threadIdx.x;
    if (i < n4) {
        float4 z = {0.f, 0.f, 0.f, 0.f};
        p[i] = z;
    }
}

// ---------------------------------------------------------------------------
// Kernel 3: per-edge gather + scatter-add. One wave32 per edge: 64 columns,
// 2 float atomics per lane; lane 0 bumps the (float) in-degree.
// ---------------------------------------------------------------------------
__global__ void __launch_bounds__(256)
scatter_edges(const float* __restrict__ h, const int* __restrict__ ei,
              const int* __restrict__ et,
              float* __restrict__ agg, float* __restrict__ cnt) {
    const int wave = threadIdx.x >> 5;
    const int lane = threadIdx.x & 31;
    const int e = blockIdx.x * 8 + wave;
    if (e >= N_EDGES) return;
    const int s = ei[e];            // edge_index[0][e] = src
    const int d = ei[N_EDGES + e];  // edge_index[1][e] = dst
    const int r = et[e];
    const float2 v = *(const float2*)(h + ((size_t)r * N_NODES + s) * F_OUT + lane * 2);
    float* ap = agg + (size_t)d * F_OUT + lane * 2;
    atomicAdd(ap + 0, v.x);
    atomicAdd(ap + 1, v.y);
    if (lane == 0) atomicAdd(&cnt[d], 1.0f);
}

// ---------------------------------------------------------------------------
// Kernel 4: out += agg / max(cnt, 1)   (float4 per thread)
// ---------------------------------------------------------------------------
__global__ void finalize(const float* __restrict__ agg, const float* __restrict__ cnt,
                         float* __restrict__ out) {
    int i = blockIdx.x * blockDim.x + threadIdx.x;  // over N_NODES*F_OUT/4
    if (i >= N_NODES * (F_OUT / 4)) return;
    const int node = i / (F_OUT / 4);
    const float c = fmaxf(cnt[node], 1.0f);
    float4 a = ((const float4*)agg)[i];
    float4 o = ((float4*)out)[i];
    o.x += a.x / c;
    o.y += a.y / c;
    o.z += a.z / c;
    o.w += a.w / c;
    ((float4*)out)[i] = o;
}

// ---------------------------------------------------------------------------
extern "C" void kernel_launch(void* const* d_in, const int* in_sizes, int n_in,
                              void* d_out, int out_size, void* d_ws, size_t ws_size,
                              hipStream_t stream) {
    const float* x     = (const float*)d_in[0];
    const int*   ei    = (const int*)d_in[1];
    const int*   et    = (const int*)d_in[2];
    const float* basis = (const float*)d_in[3];
    const float* att   = (const float*)d_in[4];
    const float* root  = (const float*)d_in[5];
    const float* bias  = (const float*)d_in[6];
    float* out = (float*)d_out;

    // workspace layout (floats): P | h[2,N,64] | agg[N,64] | cnt[N]
    float* P   = (float*)d_ws;
    float* h   = P + W_FLOATS;
    float* agg = h + (size_t)2 * N_NODES * F_OUT;
    float* cnt = agg + (size_t)N_NODES * F_OUT;

    build_weights<<<(W_FLOATS + 255) / 256, 256, 0, stream>>>(basis, att, root, P);

    const int n4 = (N_NODES * F_OUT + N_NODES) / 4;  // agg+cnt contiguous
    zero_ws<<<(n4 + 255) / 256, 256, 0, stream>>>((float4*)agg, n4);

    gemm_wmma<<<(N_TILES + WPB - 1) / WPB, 32 * WPB,
                W_FLOATS * sizeof(float), stream>>>(x, P, bias, h, out);

    scatter_edges<<<N_EDGES / 8, 256, 0, stream>>>(h, ei, et, agg, cnt);

    finalize<<<(N_NODES * (F_OUT / 4) + 255) / 256, 256, 0, stream>>>(agg, cnt, out);
}